// CenterLoss_86775519248903
// MI455X (gfx1250) — compile-verified
//
#include <hip/hip_runtime.h>
#include <math.h>

// ---------------------------------------------------------------------------
// CenterLoss for MI455X (gfx1250, wave32).
// Memory-bound (~1 GB traffic -> ~45us @ 23.3 TB/s); the N x 8 x 256 GEMM
// part uses V_WMMA_F32_16X16X4_F32 (exact f32 math, 64 wmma per 16-row tile),
// with tile staging via GLOBAL_LOAD_ASYNC_TO_LDS_B128 when available.
// All reductions are fixed-order (no float atomics) => deterministic output.
// ---------------------------------------------------------------------------

typedef float v2f __attribute__((ext_vector_type(2)));
typedef float v8f __attribute__((ext_vector_type(8)));
typedef int   v4i __attribute__((ext_vector_type(4)));

#define D 256
#define C 8

#if defined(__gfx1250__) && __has_builtin(__builtin_amdgcn_global_load_async_to_lds_b128)
#define USE_ASYNC_LDS 1
typedef __attribute__((address_space(1))) v4i as1_v4i;   // global int4
typedef __attribute__((address_space(3))) v4i as3_v4i;   // LDS int4
#else
#define USE_ASYNC_LDS 0
#endif

__device__ __forceinline__ void wait_asynccnt0() {
#if __has_builtin(__builtin_amdgcn_s_wait_asynccnt)
    __builtin_amdgcn_s_wait_asynccnt(0);
#else
    asm volatile("s_wait_asynccnt 0x0" ::: "memory");
#endif
}

// ---------------- Phase 1: per-class sums + counts (deterministic) ----------
// 128-thread blocks (4 waves). Each wave owns a contiguous row chunk and
// accumulates 8 classes x 8 cols per lane in registers (constant indices via
// switch -> no scratch). Fixed-order wave merge in LDS, one partial per block.
#define P1_BLOCKS 512
#define P1_WAVES  4
#define P1_STRIDE (C * D + C)   // 2048 sums + 8 counts per block partial

__global__ __launch_bounds__(128) void
CenterLoss_p1_classsum(const float* __restrict__ f, const long long* __restrict__ tg,
                       float* __restrict__ part, int n) {
    __shared__ float wacc[P1_WAVES][C * D];   // 32 KB
    __shared__ float wcnt[P1_WAVES][C];

    const int lane = threadIdx.x & 31;
    const int wave = threadIdx.x >> 5;

    float acc[C][8];
    float cnt[C];
#pragma unroll
    for (int c = 0; c < C; ++c) {
        cnt[c] = 0.f;
#pragma unroll
        for (int i = 0; i < 8; ++i) acc[c][i] = 0.f;
    }

    const int wavesTotal = gridDim.x * P1_WAVES;
    const int gw = blockIdx.x * P1_WAVES + wave;
    const int per = (n + wavesTotal - 1) / wavesTotal;
    int r0 = gw * per;
    int r1 = r0 + per; if (r1 > n) r1 = n;

    for (int r = r0; r < r1; ++r) {
        const int cls = (int)tg[r];
        const float4* row = (const float4*)(f + (size_t)r * D);
        float4 a = row[lane];        // cols 4*lane .. 4*lane+3
        float4 b = row[lane + 32];   // cols 128+4*lane ..
#define ACC_CASE(CI) case CI: \
        acc[CI][0]+=a.x; acc[CI][1]+=a.y; acc[CI][2]+=a.z; acc[CI][3]+=a.w; \
        acc[CI][4]+=b.x; acc[CI][5]+=b.y; acc[CI][6]+=b.z; acc[CI][7]+=b.w; \
        cnt[CI]+=1.f; break;
        switch (cls) {
            ACC_CASE(0) ACC_CASE(1) ACC_CASE(2) ACC_CASE(3)
            ACC_CASE(4) ACC_CASE(5) ACC_CASE(6) ACC_CASE(7)
            default: break;
        }
#undef ACC_CASE
    }

#pragma unroll
    for (int c = 0; c < C; ++c) {
#pragma unroll
        for (int i = 0; i < 4; ++i) {
            wacc[wave][c * D + 4 * lane + i]       = acc[c][i];
            wacc[wave][c * D + 128 + 4 * lane + i] = acc[c][4 + i];
        }
        if (lane == 0) wcnt[wave][c] = cnt[c];
    }
    __syncthreads();

    float* dst = part + (size_t)blockIdx.x * P1_STRIDE;
    for (int i = threadIdx.x; i < C * D; i += blockDim.x) {
        float s = 0.f;
#pragma unroll
        for (int w = 0; w < P1_WAVES; ++w) s += wacc[w][i];   // fixed order
        dst[i] = s;
    }
    if (threadIdx.x < C) {
        float s = 0.f;
#pragma unroll
        for (int w = 0; w < P1_WAVES; ++w) s += wcnt[w][threadIdx.x];
        dst[C * D + threadIdx.x] = s;
    }
}

// ---------------- Phase 2: centers, counts, ||c||^2 (1 block) ---------------
__global__ __launch_bounds__(256) void
CenterLoss_p2_centers(const float* __restrict__ part, int nblk,
                      float* __restrict__ centers, float* __restrict__ counts,
                      float* __restrict__ c2) {
    __shared__ float cl[C];
    __shared__ float cen[C][D];
    if (threadIdx.x < C) {
        float s = 0.f;
        for (int b = 0; b < nblk; ++b) s += part[(size_t)b * P1_STRIDE + C * D + threadIdx.x];
        counts[threadIdx.x] = s;
        cl[threadIdx.x] = s;
    }
    __syncthreads();
    const int t = threadIdx.x;   // column 0..255
    for (int c = 0; c < C; ++c) {
        float s = 0.f;
        for (int b = 0; b < nblk; ++b) s += part[(size_t)b * P1_STRIDE + c * D + t];
        float v = s / fmaxf(cl[c], 1.f);
        cen[c][t] = v;
        centers[c * D + t] = v;
    }
    __syncthreads();
    if (threadIdx.x < C) {
        float s = 0.f;
        for (int i = 0; i < D; ++i) { float v = cen[threadIdx.x][i]; s += v * v; }
        c2[threadIdx.x] = s;
    }
}

// ---------------- Phase 3: WMMA distance + row-min + partial sum ------------
// 64-thread blocks (2 waves), 16 rows per wave. LDS rows padded to 260 floats
// so the 16-lane A-fragment ds_load_b64 is bank-conflict free (260 % 64 == 4).
// Tile staging uses async global->LDS b128 (no VGPR round trip) when present.
#define P3_WAVES 2
#define ROWPAD   260

__global__ __launch_bounds__(64) void
CenterLoss_p3_dist(const float* __restrict__ f, const float* __restrict__ centers,
                   const float* __restrict__ counts, const float* __restrict__ c2,
                   float* __restrict__ part, int n) {
    __shared__ float tile[P3_WAVES][16][ROWPAD];  // ~33.3 KB
    __shared__ float cpad[16][ROWPAD];            // ~16.6 KB, rows 8..15 zero
    __shared__ float c2l[16];
    __shared__ float cntl[C];
    __shared__ float f2l[P3_WAVES][16];
    __shared__ float wsum[P3_WAVES];

    const int lane = threadIdx.x & 31;
    const int wave = threadIdx.x >> 5;

    // Stage 16x256 feature tile (coalesced b128 granularity).
    const size_t rowbase = (size_t)blockIdx.x * (P3_WAVES * 16) + (size_t)wave * 16;
    const float4* src = (const float4*)(f + rowbase * D);
#if USE_ASYNC_LDS
#pragma unroll
    for (int j = 0; j < 32; ++j) {
        int t = j * 32 + lane;              // 0..1023 = 16 rows x 64 float4
        int rr = t >> 6, cc = t & 63;
        __builtin_amdgcn_global_load_async_to_lds_b128(
            (as1_v4i*)(src + t), (as3_v4i*)&tile[wave][rr][cc * 4], 0, 0);
    }
#else
#pragma unroll
    for (int j = 0; j < 32; ++j) {
        int t = j * 32 + lane;
        int rr = t >> 6, cc = t & 63;
        *(float4*)&tile[wave][rr][cc * 4] = src[t];
    }
#endif

    // Stage zero-padded B (centers^T columns) + scalars while async is in flight.
    for (int i = threadIdx.x; i < 16 * ROWPAD; i += blockDim.x) {
        int rr = i / ROWPAD, cc = i - rr * ROWPAD;
        cpad[rr][cc] = (rr < C && cc < D) ? centers[rr * D + cc] : 0.f;
    }
    if (threadIdx.x < C)  { c2l[threadIdx.x] = c2[threadIdx.x]; cntl[threadIdx.x] = counts[threadIdx.x]; }
    if (threadIdx.x >= C && threadIdx.x < 16) c2l[threadIdx.x] = 0.f;

#if USE_ASYNC_LDS
    wait_asynccnt0();
#endif
    __syncthreads();

    // ||f||^2 per row, deterministic: lane (l&15) sums half a row, xor-16 add.
    {
        const int rr = lane & 15, h = lane >> 4;
        const float4* rp = (const float4*)&tile[wave][rr][h * 128];
        float a = 0.f;
#pragma unroll
        for (int i = 0; i < 32; ++i) { float4 v = rp[i]; a += v.x*v.x + v.y*v.y + v.z*v.z + v.w*v.w; }
        a += __shfl_xor(a, 16);
        if (h == 0) f2l[wave][rr] = a;
    }
    __syncthreads();

    // D = A(16x256) x B(256x16) via 64 x V_WMMA_F32_16X16X4_F32.
    // A frag: lane -> row (lane&15), K pair at k + 2*(lane>>4).
    // B frag: lane -> col (lane&15) of centers^T = row of cpad; same K pair.
    const int arow = lane & 15;
    const int koff = (lane >> 4) * 2;
    const float* ap = &tile[wave][arow][0];
    const float* bp = &cpad[lane & 15][0];
    v8f acc = {};
#pragma unroll
    for (int kb = 0; kb < D; kb += 4) {
        v2f a = *(const v2f*)(ap + kb + koff);
        v2f b = *(const v2f*)(bp + kb + koff);
        acc = __builtin_amdgcn_wmma_f32_16x16x4_f32(false, a, false, b,
                                                    (short)0, acc, false, false);
    }

    // Epilogue: d2 -> dist -> min over classes (width-16 shuffle) -> sum.
    const int n16 = lane & 15;
    const int h   = lane >> 4;
    const bool valid = (n16 < C) && (cntl[n16 & (C - 1)] > 0.f);
    const float c2v = c2l[n16];
    float psum = 0.f;
#pragma unroll
    for (int r = 0; r < 8; ++r) {
        const int m = r + 8 * h;                       // C/D layout: VGPR r -> rows {r, r+8}
        float f2m = f2l[wave][m];
        float d2  = f2m + c2v - 2.f * acc[r];
        float dist = valid ? sqrtf(fmaxf(d2, 0.f)) : __builtin_inff();
#pragma unroll
        for (int mm = 8; mm >= 1; mm >>= 1) dist = fminf(dist, __shfl_xor(dist, mm, 16));
        if (n16 == 0) psum += dist;                    // lanes 0 (row r) and 16 (row r+8)
    }
#pragma unroll
    for (int mm = 16; mm >= 1; mm >>= 1) psum += __shfl_xor(psum, mm);
    if (lane == 0) wsum[wave] = psum;
    __syncthreads();
    if (threadIdx.x == 0) {
        float s = 0.f;
#pragma unroll
        for (int w = 0; w < P3_WAVES; ++w) s += wsum[w];
        part[blockIdx.x] = s;
    }
}

// ---------------- Phase 3b: scalar tail (rows not covered by tiles) ---------
__global__ __launch_bounds__(256) void
CenterLoss_p3_tail(const float* __restrict__ f, const float* __restrict__ centers,
                   const float* __restrict__ counts, const float* __restrict__ c2,
                   float* __restrict__ partslot, int rbase, int n) {
    __shared__ float s[256];
    float a = 0.f;
    for (int r = rbase + threadIdx.x; r < n; r += 256) {
        const float* row = f + (size_t)r * D;
        float f2 = 0.f;
        for (int i = 0; i < D; ++i) f2 += row[i] * row[i];
        float best = __builtin_inff();
        for (int c = 0; c < C; ++c) {
            if (counts[c] > 0.f) {
                float dot = 0.f;
                for (int i = 0; i < D; ++i) dot += row[i] * centers[c * D + i];
                float d2 = f2 + c2[c] - 2.f * dot;
                best = fminf(best, sqrtf(fmaxf(d2, 0.f)));
            }
        }
        a += best;
    }
    s[threadIdx.x] = a;
    __syncthreads();
    for (int st = 128; st > 0; st >>= 1) {
        if (threadIdx.x < st) s[threadIdx.x] += s[threadIdx.x + st];
        __syncthreads();
    }
    if (threadIdx.x == 0) partslot[0] = s[0];
}

// ---------------- Phase 4: final fixed-order reduction ----------------------
__global__ __launch_bounds__(256) void
CenterLoss_p4_reduce(const float* __restrict__ part, int npart,
                     float* __restrict__ out, float invN) {
    __shared__ float s[256];
    float a = 0.f;
    for (int i = threadIdx.x; i < npart; i += 256) a += part[i];   // fixed order
    s[threadIdx.x] = a;
    __syncthreads();
    for (int st = 128; st > 0; st >>= 1) {
        if (threadIdx.x < st) s[threadIdx.x] += s[threadIdx.x + st];
        __syncthreads();
    }
    if (threadIdx.x == 0) out[0] = s[0] * invN;
}

// ---------------------------------------------------------------------------
extern "C" void kernel_launch(void* const* d_in, const int* in_sizes, int n_in,
                              void* d_out, int out_size, void* d_ws, size_t ws_size,
                              hipStream_t stream) {
    const float*     feats = (const float*)d_in[0];
    const long long* tg    = (const long long*)d_in[1];
    float*           out   = (float*)d_out;
    const int n = in_sizes[1];                 // number of rows (targets count)

    // Workspace layout (floats); total ~4.3 MB for N=524288.
    float* ws      = (float*)d_ws;
    float* part1   = ws;                                   // P1_BLOCKS * P1_STRIDE
    float* centers = part1 + (size_t)P1_BLOCKS * P1_STRIDE;
    float* counts  = centers + C * D;
    float* c2      = counts + C;
    float* part3   = c2 + C;                               // nblocks3 (+1 tail slot)

    const int rowsPerBlock3 = P3_WAVES * 16;               // 32
    const int nblocks3 = n / rowsPerBlock3;
    const int tailBase = nblocks3 * rowsPerBlock3;
    const int hasTail  = (tailBase < n) ? 1 : 0;

    CenterLoss_p1_classsum<<<P1_BLOCKS, 128, 0, stream>>>(feats, tg, part1, n);
    CenterLoss_p2_centers <<<1, 256, 0, stream>>>(part1, P1_BLOCKS, centers, counts, c2);
    if (nblocks3 > 0)
        CenterLoss_p3_dist<<<nblocks3, P3_WAVES * 32, 0, stream>>>(feats, centers, counts, c2,
                                                                   part3, n);
    if (hasTail)
        CenterLoss_p3_tail<<<1, 256, 0, stream>>>(feats, centers, counts, c2,
                                                  part3 + nblocks3, tailBase, n);
    CenterLoss_p4_reduce<<<1, 256, 0, stream>>>(part3, nblocks3 + hasTail, out,
                                                1.0f / (float)n);
}